// KNN_42571715838433
// MI455X (gfx1250) — compile-verified
//
#include <hip/hip_runtime.h>
#include <hip/hip_bf16.h>
#include <float.h>

typedef float v2f __attribute__((ext_vector_type(2)));
typedef float v8f __attribute__((ext_vector_type(8)));

#define TOPK    5
#define BLOCK   256
#define NBLOCKS 512

// Kernel 1: streaming WMMA squared-distance pass + per-block top-5.
// Each wave32 processes tiles of 16 embedding rows. Per K-chunk of 4,
// one non-temporal b64 load per lane yields registers that serve
// simultaneously as the A-matrix (16x4) and B-matrix (4x16) operands of
// V_WMMA_F32_16X16X4_F32 on the query-centered rows; the accumulated
// 16x16 Gram diagonal is the 16 squared distances.
__global__ __launch_bounds__(BLOCK) void knn_wmma_topk_kernel(
    const float* __restrict__ query,   // [64]
    const float* __restrict__ emb,     // [N][64] row-major
    int N,
    float* __restrict__ cand_d,        // [NBLOCKS*TOPK]
    int*   __restrict__ cand_i)        // [NBLOCKS*TOPK]
{
    const int tid          = threadIdx.x;
    const int lane         = tid & 31;
    const int wavesPerBlk  = BLOCK >> 5;
    const int globalWave   = blockIdx.x * wavesPerBlk + (tid >> 5);
    const int totalWaves   = gridDim.x * wavesPerBlk;

    const int off       = (lane < 16) ? 0 : 2;  // K sub-offset inside chunk of 4
    const int rowInTile = lane & 15;

    // Per-lane query fragments: q[c] = { x[4c+off], x[4c+off+1] }
    v2f q[16];
#pragma unroll
    for (int c = 0; c < 16; ++c)
        q[c] = *(const v2f*)(query + 4 * c + off);

    // Per-lane top-5 (ascending squared distance)
    float bd[TOPK];
    int   bi[TOPK];
#pragma unroll
    for (int j = 0; j < TOPK; ++j) { bd[j] = FLT_MAX; bi[j] = -1; }

    const int numTiles = (N + 15) >> 4;

    for (int tile = globalWave; tile < numTiles; tile += totalWaves) {
        const int row        = tile * 16 + rowInTile;
        const int rowClamped = (row < N) ? row : (N - 1);
        const float* base    = emb + (long long)rowClamped * 64 + off;

        // Phase 1: issue all 16 loads into distinct registers (clauseable).
        // Non-temporal: 512 MB read-once stream, keep it out of the caches.
        v2f t[16];
#pragma unroll
        for (int c = 0; c < 16; ++c)
            t[c] = __builtin_nontemporal_load((const v2f*)(base + 4 * c));

        // Phase 2: center by the query (packed f32 subtract)
#pragma unroll
        for (int c = 0; c < 16; ++c)
            t[c] -= q[c];

        // Phase 3: 16 WMMAs across 4 independent accumulator chains
        v8f acc0 = {0.f,0.f,0.f,0.f,0.f,0.f,0.f,0.f};
        v8f acc1 = {0.f,0.f,0.f,0.f,0.f,0.f,0.f,0.f};
        v8f acc2 = {0.f,0.f,0.f,0.f,0.f,0.f,0.f,0.f};
        v8f acc3 = {0.f,0.f,0.f,0.f,0.f,0.f,0.f,0.f};
#pragma unroll
        for (int c = 0; c < 16; c += 4) {
            acc0 = __builtin_amdgcn_wmma_f32_16x16x4_f32(
                false, t[c + 0], false, t[c + 0], (short)0, acc0, false, false);
            acc1 = __builtin_amdgcn_wmma_f32_16x16x4_f32(
                false, t[c + 1], false, t[c + 1], (short)0, acc1, false, false);
            acc2 = __builtin_amdgcn_wmma_f32_16x16x4_f32(
                false, t[c + 2], false, t[c + 2], (short)0, acc2, false, false);
            acc3 = __builtin_amdgcn_wmma_f32_16x16x4_f32(
                false, t[c + 3], false, t[c + 3], (short)0, acc3, false, false);
        }
        v8f acc = (acc0 + acc1) + (acc2 + acc3);

        // Diagonal extraction: lane m (m<8) holds D[m][m] in VGPR m,
        // lane m+16 (m>=8) holds D[m][m] in VGPR m-8.
        const int g = lane & 7;
        float s = acc[0];
        s = (g == 1) ? acc[1] : s;
        s = (g == 2) ? acc[2] : s;
        s = (g == 3) ? acc[3] : s;
        s = (g == 4) ? acc[4] : s;
        s = (g == 5) ? acc[5] : s;
        s = (g == 6) ? acc[6] : s;
        s = (g == 7) ? acc[7] : s;

        const bool valid = ((lane < 8) || (lane >= 24)) && (row < N);
        if (!valid) s = FLT_MAX;

        // Sorted insert into per-lane top-5
        if (s < bd[4]) {
            bd[4] = s; bi[4] = row;
#pragma unroll
            for (int j = 4; j > 0; --j) {
                if (bd[j] < bd[j - 1]) {
                    float td = bd[j]; bd[j] = bd[j - 1]; bd[j - 1] = td;
                    int   ti = bi[j]; bi[j] = bi[j - 1]; bi[j - 1] = ti;
                }
            }
        }
    }

    // Block-level reduction through LDS
    __shared__ float sd[BLOCK * TOPK];
    __shared__ int   si[BLOCK * TOPK];
#pragma unroll
    for (int j = 0; j < TOPK; ++j) {
        sd[tid * TOPK + j] = bd[j];
        si[tid * TOPK + j] = bi[j];
    }
    __syncthreads();

    if (tid == 0) {
        float td[TOPK]; int ti[TOPK];
#pragma unroll
        for (int j = 0; j < TOPK; ++j) { td[j] = FLT_MAX; ti[j] = -1; }
        for (int k = 0; k < BLOCK * TOPK; ++k) {
            float v = sd[k];
            if (v < td[4]) {
                td[4] = v; ti[4] = si[k];
#pragma unroll
                for (int j = 4; j > 0; --j) {
                    if (td[j] < td[j - 1]) {
                        float a = td[j]; td[j] = td[j - 1]; td[j - 1] = a;
                        int   b = ti[j]; ti[j] = ti[j - 1]; ti[j - 1] = b;
                    }
                }
            }
        }
#pragma unroll
        for (int j = 0; j < TOPK; ++j) {
            cand_d[blockIdx.x * TOPK + j] = td[j];
            cand_i[blockIdx.x * TOPK + j] = ti[j];
        }
    }
}

// Kernel 2: global top-5 over block candidates, label gather, mode
// (most frequent, smallest label on tie), write result as float.
__global__ void knn_finalize_kernel(const float* __restrict__ cand_d,
                                    const int*   __restrict__ cand_i,
                                    int ncand,
                                    const int* __restrict__ labels,
                                    float* __restrict__ out)
{
    if (threadIdx.x != 0 || blockIdx.x != 0) return;

    float td[TOPK]; int ti[TOPK];
#pragma unroll
    for (int j = 0; j < TOPK; ++j) { td[j] = FLT_MAX; ti[j] = -1; }

    for (int k = 0; k < ncand; ++k) {
        float v = cand_d[k];
        if (v < td[4]) {
            td[4] = v; ti[4] = cand_i[k];
#pragma unroll
            for (int j = 4; j > 0; --j) {
                if (td[j] < td[j - 1]) {
                    float a = td[j]; td[j] = td[j - 1]; td[j - 1] = a;
                    int   b = ti[j]; ti[j] = ti[j - 1]; ti[j - 1] = b;
                }
            }
        }
    }

    int lab[TOPK];
#pragma unroll
    for (int j = 0; j < TOPK; ++j) {
        int idx = (ti[j] >= 0) ? ti[j] : 0;
        lab[j] = labels[idx];
    }

    int bestLab = lab[0];
    int bestCnt = 0;
#pragma unroll
    for (int i = 0; i < TOPK; ++i) {
        int cnt = 0;
#pragma unroll
        for (int j = 0; j < TOPK; ++j) cnt += (lab[j] == lab[i]) ? 1 : 0;
        if (cnt > bestCnt || (cnt == bestCnt && lab[i] < bestLab)) {
            bestCnt = cnt;
            bestLab = lab[i];
        }
    }

    out[0] = (float)bestLab;
}

extern "C" void kernel_launch(void* const* d_in, const int* in_sizes, int n_in,
                              void* d_out, int out_size, void* d_ws, size_t ws_size,
                              hipStream_t stream)
{
    const float* query  = (const float*)d_in[0];  // [64]
    const float* emb    = (const float*)d_in[1];  // [N*64]
    const int*   labels = (const int*)d_in[2];    // [N]

    const int D = in_sizes[0];              // 64
    const int N = in_sizes[1] / D;          // 2,000,000

    float* out    = (float*)d_out;
    float* cand_d = (float*)d_ws;
    int*   cand_i = (int*)((char*)d_ws + (size_t)NBLOCKS * TOPK * sizeof(float));

    knn_wmma_topk_kernel<<<NBLOCKS, BLOCK, 0, stream>>>(query, emb, N, cand_d, cand_i);
    knn_finalize_kernel<<<1, 32, 0, stream>>>(cand_d, cand_i, NBLOCKS * TOPK, labels, out);
}